// GEANet_71863392796925
// MI455X (gfx1250) — compile-verified
//
#include <hip/hip_runtime.h>
#include <math.h>

typedef __attribute__((ext_vector_type(2))) float v2f;
typedef __attribute__((ext_vector_type(8))) float v8f;
typedef __attribute__((ext_vector_type(4))) unsigned int u32x4;
typedef __attribute__((ext_vector_type(4))) int i32x4;
typedef __attribute__((ext_vector_type(8))) int i32x8;

constexpr int BB = 8;
constexpr int LL = 12;
constexpr int NN = 2048;
constexpr int DD = 64;
constexpr int HH = 4;
constexpr int UU = 16;
constexpr int SS = 60;
constexpr int KTOP = 1638;                 // int(2048 * 0.8)
constexpr float INV_SQRT_L = 0.2886751345948129f;  // 1/sqrt(12)

#if defined(__HIP_DEVICE_COMPILE__) && defined(__gfx1250__) && \
    __has_builtin(__builtin_amdgcn_tensor_load_to_lds)
#define HAVE_TDM 1
#else
#define HAVE_TDM 0
#endif

// ---------------------------------------------------------------------------
// K1: Xsum[b,n,c] = sum_l node_x[b,l,n,c]
// ---------------------------------------------------------------------------
__global__ __launch_bounds__(256)
void xsum_kernel(const float* __restrict__ node_x, float* __restrict__ Xsum) {
  int idx = blockIdx.x * 256 + threadIdx.x;   // over BB*NN*DD = 1048576
  int b = idx >> 17;                          // NN*DD = 131072 = 2^17
  int rem = idx & 131071;
  const float* p = node_x + (size_t)b * (LL * NN * DD) + rem;
  float acc = 0.f;
  #pragma unroll
  for (int l = 0; l < LL; ++l) acc += p[(size_t)l * (NN * DD)];
  Xsum[idx] = acc;
}

// ---------------------------------------------------------------------------
// K2: external attention -> s[b,l,n']   (one workgroup per (b,l,h))
//   P[n,s] = x[b,l,n,h*16+:16] . U1[:,s];  softmax over n (column-wise);
//   r[n] = sum_s softmax(P)[n,s] * rowsum(U2)[s];
//   s[b,l, h*512 + n/4] = r[n]+..+r[n+3]   (faithful to the buggy reshape)
// ---------------------------------------------------------------------------
__global__ __launch_bounds__(256)
void attn_s_kernel(const float* __restrict__ node_x,
                   const float* __restrict__ U1,    // [16][60]
                   const float* __restrict__ U2,    // [60][16]
                   float* __restrict__ sbuf) {      // [BB][LL][NN]
  __shared__ float U1t[64 * 16];   // [s][u], s padded to 64
  __shared__ float colw[64];       // rowsum(U2)[s], then /colsum[s]
  __shared__ float colsum[64];
  __shared__ float rbuf[256];

  const int tid = threadIdx.x;
  const int wg  = blockIdx.x;          // BB*LL*HH = 384
  const int h = wg & 3;
  const int l = (wg >> 2) % LL;
  const int b = wg / (LL * HH);

  for (int i = tid; i < 64 * 16; i += 256) U1t[i] = 0.f;
  __syncthreads();
  for (int i = tid; i < SS * UU; i += 256) {
    int s = i >> 4, u = i & 15;
    U1t[s * 16 + u] = U1[u * SS + s];
  }
  if (tid < 64) {
    float v = 0.f;
    if (tid < SS) {
      #pragma unroll
      for (int u = 0; u < UU; ++u) v += U2[tid * UU + u];
    }
    colw[tid] = v;
    colsum[tid] = 0.f;
  }
  __syncthreads();

  const float* xbase = node_x + ((size_t)(b * LL + l) * NN) * DD + h * UU;

  // pass 1: colsum[s] = sum_n exp(P[n,s]); logits ~N(0,4) so max-free is safe
  {
    int s  = tid & 63;
    int n0 = tid >> 6;                 // 0..3
    if (s < SS) {
      const float* u1row = &U1t[s * 16];
      float local = 0.f;
      for (int n = n0; n < NN; n += 4) {
        const float* xr = xbase + (size_t)n * DD;
        float p = 0.f;
        #pragma unroll
        for (int u = 0; u < UU; ++u) p += xr[u] * u1row[u];
        local += __expf(p);
      }
      atomicAdd(&colsum[s], local);
    }
  }
  __syncthreads();
  if (tid < 64) colw[tid] = (tid < SS) ? colw[tid] / colsum[tid] : 0.f;
  __syncthreads();

  // pass 2: r[n], then group-of-4 fold into s
  float* sout = sbuf + (size_t)(b * LL + l) * NN;
  for (int iter = 0; iter < 8; ++iter) {
    int n = iter * 256 + tid;
    const float* xr = xbase + (size_t)n * DD;
    float x[UU];
    #pragma unroll
    for (int u = 0; u < UU; ++u) x[u] = xr[u];
    float r = 0.f;
    for (int s = 0; s < SS; ++s) {
      const float* u1row = &U1t[s * 16];
      float p = 0.f;
      #pragma unroll
      for (int u = 0; u < UU; ++u) p += x[u] * u1row[u];
      r += __expf(p) * colw[s];
    }
    rbuf[tid] = r;
    __syncthreads();
    if (tid < 64) {
      float v = rbuf[tid * 4] + rbuf[tid * 4 + 1] + rbuf[tid * 4 + 2] + rbuf[tid * 4 + 3];
      sout[h * 512 + iter * 64 + tid] = v;
    }
    __syncthreads();
  }
}

// ---------------------------------------------------------------------------
// K3: adj1 relu-logits, LDS-tiled WMMA + Tensor Data Mover staging.
//   L1[b,n,m] = relu(inv_sqrt_l * Xsum[b,n,:] . emb[m,:])
// Macro-tile: 64 rows x 128 cols per workgroup. TDM loads A (64x64) and
// B (128x64) tiles into LDS with pad_enable (row stride 66 DWORDs ->
// bank-conflict-free fragment reads: bank = (2*row + k) mod 64).
// Each of 8 waves owns a 16x64 slab: 4 accumulators x 16 k-steps = 64 WMMAs.
// ---------------------------------------------------------------------------
#define LDS_STRIDE 66
#define A_ROWS 64
#define B_ROWS 128

#if HAVE_TDM
__device__ __forceinline__ void tdm_load_2d(unsigned lds_off,
                                            const void* gaddr,
                                            unsigned tile_rows) {
  unsigned long long ga = (unsigned long long)(size_t)gaddr;
  u32x4 g0;
  g0.x = 1u;                                  // count=1 valid user descriptor
  g0.y = lds_off;                             // lds_addr (bytes)
  g0.z = (unsigned)(ga & 0xFFFFFFFFu);        // global_addr[31:0]
  g0.w = (unsigned)((ga >> 32) & 0x01FFFFFFu) // global_addr[56:32]
         | (2u << 30);                        // type = 2 ("image")
  i32x8 g1;
  // data_size=2 (4B) | pad_enable | pad_interval=64 DWORDs | pad_amount=2 DW
  g1[0] = (int)((2u << 16) | (1u << 20) | (5u << 22) | (1u << 25));
  g1[1] = (int)(64u << 16);        // tensor_dim0 = 64 (bits 79:48, low part)
  g1[2] = (int)(tile_rows << 16);  // tensor_dim0 hi=0 | tensor_dim1 low
  g1[3] = (int)(64u << 16);        // tensor_dim1 hi=0 | tile_dim0 = 64
  g1[4] = (int)tile_rows;          // tile_dim1 | tile_dim2=0
  g1[5] = 64;                      // tensor_dim0_stride = 64 (low 32)
  g1[6] = 0;                       // stride0 hi | stride1 low
  g1[7] = 0;                       // stride1 hi (2D: unused)
  i32x4 z4 = {0, 0, 0, 0};
#if __clang_major__ >= 23
  i32x8 z8 = {0, 0, 0, 0, 0, 0, 0, 0};
  __builtin_amdgcn_tensor_load_to_lds(g0, g1, z4, z4, z8, 0);
#else
  __builtin_amdgcn_tensor_load_to_lds(g0, g1, z4, z4, 0);
#endif
}
#endif

__global__ __launch_bounds__(256)
void adj1_wmma_kernel(const float* __restrict__ Xsum,   // [BB][NN][64]
                      const float* __restrict__ emb,    // [NN][64]
                      float* __restrict__ L1) {         // [BB][NN][NN]
  __shared__ float smem[(A_ROWS + B_ROWS) * LDS_STRIDE];  // 50.7 KB

  const int tid  = threadIdx.x;
  const int wid  = tid >> 5;
  const int lane = tid & 31;

  const int t  = blockIdx.x;        // 8 * 32 * 16 = 4096 macro-tiles
  const int mc = t & 15;            // 128-col macro index
  const int ar = (t >> 4) & 31;     // 64-row macro index
  const int bb = t >> 9;            // batch

  const float* Ag = Xsum + ((size_t)bb * NN + ar * A_ROWS) * DD;
  const float* Bg = emb + (size_t)mc * B_ROWS * DD;
  float* As = smem;
  float* Bs = smem + A_ROWS * LDS_STRIDE;

#if HAVE_TDM
  if (wid == 0) {
    unsigned base = (unsigned)(size_t)(void*)smem;   // LDS aperture low bits
    tdm_load_2d(base, Ag, A_ROWS);
    tdm_load_2d(base + A_ROWS * LDS_STRIDE * 4, Bg, B_ROWS);
    __builtin_amdgcn_s_wait_tensorcnt(0);
  }
  __syncthreads();
#else
  for (int i = tid; i < A_ROWS * DD; i += 256) {
    int r = i >> 6, c = i & 63;
    As[r * LDS_STRIDE + c] = Ag[(size_t)r * DD + c];
  }
  for (int i = tid; i < B_ROWS * DD; i += 256) {
    int r = i >> 6, c = i & 63;
    Bs[r * LDS_STRIDE + c] = Bg[(size_t)r * DD + c];
  }
  __syncthreads();
#endif

  const int lane16 = lane & 15;
  const int khalf  = (lane >> 4) << 1;     // 0 or 2
  const int rowblk = wid >> 1;             // 0..3  (16-row slab)
  const int colgrp = wid & 1;              // 0..1  (64-col half)

  const float* arow = As + (rowblk * 16 + lane16) * LDS_STRIDE;
  v8f acc[4] = {};
  #pragma unroll
  for (int kk = 0; kk < 16; ++kk) {
    const int k0 = kk * 4 + khalf;
    v2f a;
    a.x = arow[k0]; a.y = arow[k0 + 1];
    #pragma unroll
    for (int mt = 0; mt < 4; ++mt) {
      const float* brow = Bs + (colgrp * 64 + mt * 16 + lane16) * LDS_STRIDE;
      v2f bf;
      bf.x = brow[k0]; bf.y = brow[k0 + 1];
      acc[mt] = __builtin_amdgcn_wmma_f32_16x16x4_f32(false, a, false, bf,
                                                      (short)0, acc[mt],
                                                      false, false);
    }
  }

  const int rbase = ar * A_ROWS + rowblk * 16 + ((lane >> 4) << 3);
  float* outb = L1 + (size_t)bb * NN * NN;
  #pragma unroll
  for (int mt = 0; mt < 4; ++mt) {
    const int c = mc * B_ROWS + colgrp * 64 + mt * 16 + lane16;
    #pragma unroll
    for (int i = 0; i < 8; ++i) {
      float v = fmaxf(acc[mt][i] * INV_SQRT_L, 0.f);
      outb[(size_t)(rbase + i) * NN + c] = v;
    }
  }
}

// ---------------------------------------------------------------------------
// K3b: adj2 relu-logits via WMMA, K=12 (3 k-steps).
//   L2[b,n,m] = relu(inv_sqrt_l * sum_c s[b,c,n]*s[b,c,m])
// A fragment rows are sT (stride-NN loads, L2-resident); B rows contiguous.
// ---------------------------------------------------------------------------
__global__ __launch_bounds__(256)
void adj2_wmma_kernel(const float* __restrict__ sbuf,   // [BB][LL][NN]
                      float* __restrict__ L2buf) {      // [BB][NN][NN]
  const int tid  = threadIdx.x;
  const int wid  = tid >> 5;
  const int lane = tid & 31;
  const int t   = blockIdx.x * 8 + wid;    // tile id over BB*128*128
  const int bb  = t >> 14;
  const int rem = t & 16383;
  const int nt  = rem >> 7;
  const int mt  = rem & 127;

  const int lane16 = lane & 15;
  const int khalf  = (lane >> 4) << 1;     // 0 or 2

  const float* sb = sbuf + (size_t)bb * LL * NN;
  const int nrow = nt * 16 + lane16;
  const int mcol = mt * 16 + lane16;

  v8f acc = {};
  #pragma unroll
  for (int kk = 0; kk < 3; ++kk) {
    const int k0 = kk * 4 + khalf;
    v2f a, bf;
    a.x  = sb[(size_t)k0 * NN + nrow];
    a.y  = sb[(size_t)(k0 + 1) * NN + nrow];
    bf.x = sb[(size_t)k0 * NN + mcol];
    bf.y = sb[(size_t)(k0 + 1) * NN + mcol];
    acc = __builtin_amdgcn_wmma_f32_16x16x4_f32(false, a, false, bf,
                                                (short)0, acc, false, false);
  }

  const int rbase = nt * 16 + ((lane >> 4) << 3);
  float* outb = L2buf + (size_t)bb * NN * NN;
  #pragma unroll
  for (int i = 0; i < 8; ++i) {
    float v = fmaxf(acc[i] * INV_SQRT_L, 0.f);
    outb[(size_t)(rbase + i) * NN + mcol] = v;
  }
}

// ---------------------------------------------------------------------------
// K4: per-row fused softmax(adj1) / softmax(adj2) / fuse / softmax / top-k.
// One workgroup per (b, n); both 2048-rows live in LDS. Reductions use
// wave32 shfl_xor butterflies + one 8-partial LDS combine.
// ---------------------------------------------------------------------------
__device__ __forceinline__ float wred_sum(float v) {
  #pragma unroll
  for (int off = 16; off > 0; off >>= 1) v += __shfl_xor(v, off, 32);
  return v;
}
__device__ __forceinline__ float wred_max(float v) {
  #pragma unroll
  for (int off = 16; off > 0; off >>= 1) v = fmaxf(v, __shfl_xor(v, off, 32));
  return v;
}
__device__ __forceinline__ float block_sum(float v, float* red, int tid) {
  v = wred_sum(v);
  if ((tid & 31) == 0) red[tid >> 5] = v;
  __syncthreads();
  float r = red[0] + red[1] + red[2] + red[3] +
            red[4] + red[5] + red[6] + red[7];
  __syncthreads();
  return r;
}
__device__ __forceinline__ float block_max(float v, float* red, int tid) {
  v = wred_max(v);
  if ((tid & 31) == 0) red[tid >> 5] = v;
  __syncthreads();
  float r = fmaxf(fmaxf(fmaxf(red[0], red[1]), fmaxf(red[2], red[3])),
                  fmaxf(fmaxf(red[4], red[5]), fmaxf(red[6], red[7])));
  __syncthreads();
  return r;
}

__global__ __launch_bounds__(256)
void fuse_topk_kernel(const float* __restrict__ L1,
                      const float* __restrict__ L2buf,
                      const float* __restrict__ fcw,    // [2]
                      const float* __restrict__ fcb,    // [1]
                      float* __restrict__ out) {
  __shared__ __align__(16) float R1[NN];
  __shared__ __align__(16) float R2[NN];
  __shared__ float red[8];
  __shared__ int   scan[256];

  const int tid = threadIdx.x;
  const int b = blockIdx.x >> 11;
  const int n = blockIdx.x & 2047;

  const float w0 = fcw[0], w1 = fcw[1], bias = fcb[0];

  // vectorized row loads (b128)
  {
    const float4* l1v = (const float4*)(L1 + ((size_t)b * NN + n) * NN);
    const float4* l2v = (const float4*)(L2buf + ((size_t)b * NN + n) * NN);
    float4* r1v = (float4*)R1;
    float4* r2v = (float4*)R2;
    r1v[tid * 2]     = l1v[tid * 2];
    r1v[tid * 2 + 1] = l1v[tid * 2 + 1];
    r2v[tid * 2]     = l2v[tid * 2];
    r2v[tid * 2 + 1] = l2v[tid * 2 + 1];
  }
  __syncthreads();

  // softmax #1 stats (adj1 row)
  float lm = -3.4e38f;
  #pragma unroll
  for (int j = 0; j < 8; ++j) lm = fmaxf(lm, R1[tid * 8 + j]);
  const float M1 = block_max(lm, red, tid);
  float ls = 0.f;
  #pragma unroll
  for (int j = 0; j < 8; ++j) ls += __expf(R1[tid * 8 + j] - M1);
  const float Z1 = block_sum(ls, red, tid);

  // softmax #2 stats (adj2 row)
  float lm2 = -3.4e38f;
  #pragma unroll
  for (int j = 0; j < 8; ++j) lm2 = fmaxf(lm2, R2[tid * 8 + j]);
  const float M2 = block_max(lm2, red, tid);
  float ls2 = 0.f;
  #pragma unroll
  for (int j = 0; j < 8; ++j) ls2 += __expf(R2[tid * 8 + j] - M2);
  const float Z2 = block_sum(ls2, red, tid);

  // fuse, then softmax #3 in-place in R1
  float lf = -3.4e38f;
  #pragma unroll
  for (int j = 0; j < 8; ++j) {
    int m = tid * 8 + j;
    float p1 = __expf(R1[m] - M1) / Z1;
    float p2 = __expf(R2[m] - M2) / Z2;
    float f = w0 * p1 + w1 * p2 + bias;
    R1[m] = f;
    lf = fmaxf(lf, f);
  }
  const float Mf = block_max(lf, red, tid);
  float lsf = 0.f;
  #pragma unroll
  for (int j = 0; j < 8; ++j) lsf += __expf(R1[tid * 8 + j] - Mf);
  const float Zf = block_sum(lsf, red, tid);
  #pragma unroll
  for (int j = 0; j < 8; ++j) {
    int m = tid * 8 + j;
    R1[m] = __expf(R1[m] - Mf) / Zf;   // p[m] in (0,1]: positive floats
  }
  __syncthreads();

  // top-k threshold: largest T (uint bits; monotone for positive floats)
  // with count(p >= T) >= KTOP.  p <= 1.0 => bits < 2^30: start at bit 29.
  unsigned cur = 0u;
  for (int bit = 29; bit >= 0; --bit) {
    unsigned cand = cur | (1u << bit);
    float c = 0.f;
    #pragma unroll
    for (int j = 0; j < 8; ++j)
      c += (__float_as_uint(R1[tid * 8 + j]) >= cand) ? 1.f : 0.f;
    if (block_sum(c, red, tid) >= (float)KTOP) cur = cand;
  }
  const unsigned T = cur;

  float cg = 0.f;
  int local_eq = 0;
  #pragma unroll
  for (int j = 0; j < 8; ++j) {
    unsigned u = __float_as_uint(R1[tid * 8 + j]);
    if (u > T) cg += 1.f;
    if (u == T) ++local_eq;
  }
  const int cnt_gt = (int)block_sum(cg, red, tid);
  const int n_take = KTOP - cnt_gt;   // ties kept lowest-index-first

  // exclusive prefix over per-thread tie counts (thread t owns contiguous
  // columns [8t, 8t+8) -> rank order == index order)
  scan[tid] = local_eq;
  __syncthreads();
  for (int off = 1; off < 256; off <<= 1) {
    int v = (tid >= off) ? scan[tid - off] : 0;
    __syncthreads();
    scan[tid] += v;
    __syncthreads();
  }
  int rank = scan[tid] - local_eq;

  // masked output, vectorized stores
  float4 o[2];
  float* of = (float*)o;
  #pragma unroll
  for (int j = 0; j < 8; ++j) {
    int m = tid * 8 + j;
    float p = R1[m];
    unsigned u = __float_as_uint(p);
    bool keep = (u > T) || (u == T && (rank++ < n_take));
    of[j] = keep ? p : 0.f;
  }
  float4* ov = (float4*)(out + ((size_t)b * NN + n) * NN);
  ov[tid * 2]     = o[0];
  ov[tid * 2 + 1] = o[1];
}

// ---------------------------------------------------------------------------
extern "C" void kernel_launch(void* const* d_in, const int* in_sizes, int n_in,
                              void* d_out, int out_size, void* d_ws, size_t ws_size,
                              hipStream_t stream) {
  const float* node_x = (const float*)d_in[0];
  const float* emb    = (const float*)d_in[1];
  const float* U1     = (const float*)d_in[2];
  const float* U2     = (const float*)d_in[3];
  const float* fcw    = (const float*)d_in[4];
  const float* fcb    = (const float*)d_in[5];
  float* out = (float*)d_out;

  // workspace (floats): Xsum 1,048,576 | s 196,608 | L1 33,554,432 | L2 same
  float* Xsum  = (float*)d_ws;
  float* sbuf  = Xsum + (size_t)BB * NN * DD;
  float* L1    = sbuf + (size_t)BB * LL * NN;
  float* L2buf = L1 + (size_t)BB * NN * NN;

  xsum_kernel<<<(BB * NN * DD) / 256, 256, 0, stream>>>(node_x, Xsum);
  attn_s_kernel<<<BB * LL * HH, 256, 0, stream>>>(node_x, U1, U2, sbuf);
  adj1_wmma_kernel<<<BB * 32 * 16, 256, 0, stream>>>(Xsum, emb, L1);
  adj2_wmma_kernel<<<(BB * 128 * 128) / 8, 256, 0, stream>>>(sbuf, L2buf);
  fuse_topk_kernel<<<BB * NN, 256, 0, stream>>>(L1, L2buf, fcw, fcb, out);
}